// QuantumDQN_1614907704058
// MI455X (gfx1250) — compile-verified
//
#include <hip/hip_runtime.h>

typedef float v2f __attribute__((ext_vector_type(2)));
typedef float v8f __attribute__((ext_vector_type(8)));

#define NQ   12
#define DIM  4096            // 2^12 amplitudes
#define NPAIR 2048
#define NL   3
#define TPB  256             // 8 wave32s

__global__ __launch_bounds__(TPB) void qsim_kernel(
    const float* __restrict__ inputs,     // (B, 12)
    const float* __restrict__ w_input2,   // (12)
    const float* __restrict__ w_output2,  // (2)
    const float* __restrict__ y_weights,  // (3, 12)
    const float* __restrict__ z_weights,  // (3, 12)
    float* __restrict__ out)              // (B, 2)
{
    __shared__ float2 st[DIM];                    // 32 KB state, LDS-resident
    __shared__ float  cxs[NQ],  sxs[NQ];          // RX cos/sin per wire (batch-dep)
    __shared__ float  cys[NL*NQ], sys[NL*NQ];     // RY cos/sin per layer/wire
    __shared__ float  hzs[NL*NQ];                 // RZ half-angles
    __shared__ float  red[TPB];                   // cross-wave reduction

    const int tid = threadIdx.x;
    const int b   = blockIdx.x;

    // ---- parameter preprocessing (once per workgroup) ----
    if (tid < NQ) {
        float h = 0.5f * atanf(inputs[b * NQ + tid] * w_input2[tid]);
        cxs[tid] = cosf(h);
        sxs[tid] = sinf(h);
    }
    if (tid < NL * NQ) {
        float hy = 0.5f * y_weights[tid];
        cys[tid] = cosf(hy);
        sys[tid] = sinf(hy);
        hzs[tid] = 0.5f * z_weights[tid];
    }
    // ---- |0...0> init ----
    for (int k = tid; k < DIM; k += TPB)
        st[k] = make_float2(0.0f, 0.0f);
    __syncthreads();
    if (tid == 0) st[0] = make_float2(1.0f, 0.0f);

    for (int l = 0; l < NL; ++l) {
        // ---------- RX per wire: s0' = c*s0 - i s*s1 ; s1' = -i s*s0 + c*s1 ----------
        for (int w = 0; w < NQ; ++w) {
            __syncthreads();
            const float c = cxs[w], s = sxs[w];
            const int d1 = 1 << (11 - w);
#pragma unroll
            for (int k = 0; k < NPAIR / TPB; ++k) {
                int p  = tid + k * TPB;
                int lo = p & (d1 - 1);
                int i0 = ((p - lo) << 1) | lo;
                int i1 = i0 + d1;
                float2 a0 = st[i0], a1 = st[i1];
                float2 n0, n1;
                n0.x =  c * a0.x + s * a1.y;
                n0.y =  c * a0.y - s * a1.x;
                n1.x =  s * a0.y + c * a1.x;
                n1.y = -s * a0.x + c * a1.y;
                st[i0] = n0; st[i1] = n1;
            }
        }
        // ---------- RY per wire: s0' = c*s0 - s*s1 ; s1' = s*s0 + c*s1 ----------
        for (int w = 0; w < NQ; ++w) {
            __syncthreads();
            const float c = cys[l * NQ + w], s = sys[l * NQ + w];
            const int d1 = 1 << (11 - w);
#pragma unroll
            for (int k = 0; k < NPAIR / TPB; ++k) {
                int p  = tid + k * TPB;
                int lo = p & (d1 - 1);
                int i0 = ((p - lo) << 1) | lo;
                int i1 = i0 + d1;
                float2 a0 = st[i0], a1 = st[i1];
                float2 n0, n1;
                n0.x = c * a0.x - s * a1.x;
                n0.y = c * a0.y - s * a1.y;
                n1.x = s * a0.x + c * a1.x;
                n1.y = s * a0.y + c * a1.y;
                st[i0] = n0; st[i1] = n1;
            }
        }
        // ---------- fused 12x RZ + ring-CZ diagonal (single elementwise pass) ----------
        __syncthreads();
        float hz[NQ];
#pragma unroll
        for (int w = 0; w < NQ; ++w) hz[w] = hzs[l * NQ + w];
#pragma unroll
        for (int k = 0; k < DIM / TPB; ++k) {
            int idx = tid + k * TPB;
            // total RZ phase: + h_w if wire-bit set, - h_w otherwise (wire w <-> bit 11-w)
            float ang = 0.0f;
#pragma unroll
            for (int bp = 0; bp < NQ; ++bp) {
                float h = hz[11 - bp];
                ang += ((idx >> bp) & 1) ? h : -h;
            }
            // ring CZ sign: parity of adjacent-set-bit pairs (cyclic over 12 bits)
            int rot = ((idx << 1) | (idx >> 11)) & 0xFFF;
            float sgn = (__popc(idx & rot) & 1) ? -1.0f : 1.0f;
            float sA, cA;
            sincosf(ang, &sA, &cA);
            float2 a = st[idx];
            float2 r;
            r.x = sgn * (a.x * cA - a.y * sA);
            r.y = sgn * (a.x * sA + a.y * cA);
            st[idx] = r;
        }
    }
    __syncthreads();

    // ---------- expectation values via V_WMMA_F32_16X16X4_F32 ----------
    // 64 tiles of 64 amplitudes. Within a tile, bits 11..8 are constant, so the
    // ZZ(0,1) and ZZ(2,3) eigenvalues are per-tile constants s01, s23.
    // A (16x4) = tile probabilities, B col0 = s01, col1 = s23, others 0.
    // Accumulating D over tiles, column-0 total = e0, column-1 total = e1.
    const int lane = tid & 31;
    const int wave = tid >> 5;
    const int nsel = lane & 15;      // A-row m, and B/D column index
    v8f acc = {0.f, 0.f, 0.f, 0.f, 0.f, 0.f, 0.f, 0.f};
#pragma unroll
    for (int t = 0; t < 8; ++t) {
        const int tile = wave * 8 + t;
        const int base = tile << 6;
        const float s01 = (((base >> 11) ^ (base >> 10)) & 1) ? -1.0f : 1.0f;
        const float s23 = (((base >> 9)  ^ (base >> 8))  & 1) ? -1.0f : 1.0f;
        const float bv  = (nsel == 0) ? s01 : ((nsel == 1) ? s23 : 0.0f);
        v2f bmat; bmat[0] = bv; bmat[1] = bv;   // identical rows per column
        // A layout: lane (m = lane&15), K-pair selected by lane>>4 -> consecutive idx
        const int aidx = base + nsel * 4 + ((lane >> 4) << 1);
        float2 a0 = st[aidx], a1 = st[aidx + 1];
        v2f amat;
        amat[0] = a0.x * a0.x + a0.y * a0.y;
        amat[1] = a1.x * a1.x + a1.y * a1.y;
        acc = __builtin_amdgcn_wmma_f32_16x16x4_f32(
            /*neg_a=*/false, amat, /*neg_b=*/false, bmat,
            /*c_mod=*/(short)0, acc, /*reuse_a=*/false, /*reuse_b=*/false);
    }
    // per-lane sum over the 8 D VGPRs (covers M=0..7 or 8..15 of each column)
    float tsum = acc[0] + acc[1] + acc[2] + acc[3] +
                 acc[4] + acc[5] + acc[6] + acc[7];
    red[tid] = tsum;
    __syncthreads();

    // D column n lives in lanes n (M=0..7) and n+16 (M=8..15); combine 8 waves.
    if (tid < 2) {
        float e = 0.0f;
#pragma unroll
        for (int wv = 0; wv < TPB / 32; ++wv)
            e += red[wv * 32 + tid] + red[wv * 32 + 16 + tid];
        out[b * 2 + tid] = (1.0f + e) * 0.5f * w_output2[tid];
    }
}

extern "C" void kernel_launch(void* const* d_in, const int* in_sizes, int n_in,
                              void* d_out, int out_size, void* d_ws, size_t ws_size,
                              hipStream_t stream) {
    const float* inputs    = (const float*)d_in[0];  // (B, 12)
    const float* w_input2  = (const float*)d_in[1];  // (12)
    const float* w_output2 = (const float*)d_in[2];  // (2)
    const float* y_weights = (const float*)d_in[3];  // (3, 12)
    const float* z_weights = (const float*)d_in[4];  // (3, 12)
    float* out = (float*)d_out;

    const int B = in_sizes[0] / NQ;   // 512
    qsim_kernel<<<B, TPB, 0, stream>>>(inputs, w_input2, w_output2,
                                       y_weights, z_weights, out);
}